// Contrast_7301444403840
// MI455X (gfx1250) — compile-verified
//
#include <hip/hip_runtime.h>

// Problem constants (N = 10000 = 625 * 16, D = 8)
#define NROWS 10000
#define NT    625            // 16-row tiles
#define JCHUNK 125           // j-tiles per block (NT / NCHUNK)
#define NCHUNK 5             // j-chunks (grid.y)
#define CCOLS (JCHUNK * 16)  // columns per chunk = 2000 (8 KB LDS accumulator)
#define TAU_INV 2.0f         // 1 / 0.5
#define LOSS_EPS 1e-8f
// exp(cos/tau) = exp2(cos * TAU_INV * log2(e)); fold this scale into U1 rows.
#define U1_SCALE 2.88539008177792681472f   // TAU_INV * log2(e)
#define LN2F     0.69314718055994530942f

typedef __attribute__((ext_vector_type(2))) float v2f;
typedef __attribute__((ext_vector_type(8))) float v8f;

union V8F { v8f v; float f[8]; };

#if __has_builtin(__builtin_amdgcn_exp2f)
#define EXP2(x) __builtin_amdgcn_exp2f(x)
#else
#define EXP2(x) __expf((x) * LN2F)   // still lowers to v_exp_f32
#endif

// ds_swizzle_b32, group-of-32 xor mode: offset = {xor_mask[14:10], or=0, and=0x1f}
#define SWZ_XOR(x, m) \
    __int_as_float(__builtin_amdgcn_ds_swizzle(__float_as_int(x), ((m) << 10) | 0x1f))

// ---------------------------------------------------------------------------
// Kernel 0: zero a float range (pad tile + rowsum + colsum; ws is poisoned)
// ---------------------------------------------------------------------------
__global__ void zero_kernel(float* __restrict__ p, int n) {
    int i = blockIdx.x * blockDim.x + threadIdx.x;
    if (i < n) p[i] = 0.0f;
}

// ---------------------------------------------------------------------------
// Kernel 1: per-row 8->8->8 MLP (ELU) + L2 normalize -> fp32 unit rows.
// U1 rows are additionally scaled by TAU_INV*log2(e) so the similarity pass
// can feed WMMA results straight into v_exp_f32 (exp2).
// ---------------------------------------------------------------------------
__global__ __launch_bounds__(256) void proj_norm_kernel(
    const float* __restrict__ z_mp, const float* __restrict__ z_sc,
    const float* __restrict__ W1, const float* __restrict__ b1,
    const float* __restrict__ W2, const float* __restrict__ b2,
    float* __restrict__ U1, float* __restrict__ U2)
{
    int t = blockIdx.x * blockDim.x + threadIdx.x;
    if (t >= 2 * NROWS) return;
    bool first = (t < NROWS);
    int row = first ? t : t - NROWS;
    const float* z = (first ? z_mp : z_sc) + (size_t)row * 8;

    float x[8], h[8], y[8];
#pragma unroll
    for (int k = 0; k < 8; ++k) x[k] = z[k];

#pragma unroll
    for (int j = 0; j < 8; ++j) {                 // h = ELU(z @ W1^T + b1)
        float s = b1[j];
#pragma unroll
        for (int k = 0; k < 8; ++k) s += x[k] * W1[j * 8 + k];
        h[j] = (s > 0.0f) ? s : (__expf(s) - 1.0f);
    }

    float n2 = 0.0f;
#pragma unroll
    for (int j = 0; j < 8; ++j) {                 // y = h @ W2^T + b2
        float s = b2[j];
#pragma unroll
        for (int k = 0; k < 8; ++k) s += h[k] * W2[j * 8 + k];
        y[j] = s;
        n2 += s * s;
    }
    float inv = rsqrtf(n2) * (first ? U1_SCALE : 1.0f);

    float* U = (first ? U1 : U2) + (size_t)row * 8;
    float4 lo = make_float4(y[0] * inv, y[1] * inv, y[2] * inv, y[3] * inv);
    float4 hi = make_float4(y[4] * inv, y[5] * inv, y[6] * inv, y[7] * inv);
    *(float4*)(U)     = lo;
    *(float4*)(U + 4) = hi;
}

// ---------------------------------------------------------------------------
// Kernel 2: fused similarity pass, parallel over (i-band, j-chunk).
// blockIdx.x: band of 8 i-tiles (one per wave); blockIdx.y: chunk of jcount
// j-tiles. jcount is a RUNTIME argument so the compiler cannot fully unroll
// the loop (round 3 fully unrolled the 125-trip loop and spilled to scratch).
//
// Two v_wmma_f32_16x16x4_f32 per tile (K = 0..3 then 4..7, f32 A/B so every
// lane carries real data), result fed directly into v_exp_f32. Row partials
// in registers -> global f32 atomics (NCHUNK contributors/row). Column
// partials in an 8 KB LDS accumulator: ALL 32 lanes ds_add_f32 (lanes l and
// l+16 share an address; the DS atomic unit serializes the pair — cheaper
// than a shuffle fold + EXEC-masked branch per iteration).
//
// f32 A 16x4 layout (ISA 7.12.2): lane l holds row M=(l&15); VGPR0/1 hold
// K = (l>>4)*2 and +1. B 4x16 mirrors: lane l holds col N=(l&15), same K
// pair. Each lane loads a float2 at row*8 + (l>>4)*2 (+4 for the second
// WMMA). B for tile j+1 is prefetched (zero pad tile past U2) so the load
// is unconditional and fully overlapped.
// ---------------------------------------------------------------------------
__global__ __launch_bounds__(256) void sim_tile_kernel(
    const float* __restrict__ U1, const float* __restrict__ U2,
    float* __restrict__ rowsum, float* __restrict__ colsum, int jcount)
{
    __shared__ float colacc[CCOLS];
    for (int i = threadIdx.x; i < CCOLS; i += 256) colacc[i] = 0.0f;
    __syncthreads();

    const int lane  = threadIdx.x & 31;
    const int wave  = threadIdx.x >> 5;
    const int itile = blockIdx.x * 8 + wave;
    const int jbase = blockIdx.y * jcount;        // first j-tile of this chunk

    if (itile < NT) {
        const int lrow = ((lane & 15) << 3) + ((lane >> 4) << 1);

        const float* Ap = U1 + (size_t)itile * 128 + lrow;
        v2f a0 = *(const v2f*)(Ap);
        v2f a1 = *(const v2f*)(Ap + 4);

        const float* Bp = U2 + (size_t)jbase * 128 + lrow;   // first fragment
        v2f b0 = *(const v2f*)(Bp);
        v2f b1 = *(const v2f*)(Bp + 4);

        float racc[8];
#pragma unroll
        for (int e = 0; e < 8; ++e) racc[e] = 0.0f;

#pragma unroll 2
        for (int jj = 0; jj < jcount; ++jj) {
            // Prefetch tile jbase+jj+1 (last chunk's final prefetch hits the
            // zeroed pad tile at index NT — always a valid address)
            const float* Bn = U2 + (size_t)(jbase + jj + 1) * 128 + lrow;
            v2f nb0 = *(const v2f*)(Bn);
            v2f nb1 = *(const v2f*)(Bn + 4);

            v8f c = {};
            c = __builtin_amdgcn_wmma_f32_16x16x4_f32(
                    false, a0, false, b0, (short)0, c, false, false);
            c = __builtin_amdgcn_wmma_f32_16x16x4_f32(
                    false, a1, false, b1, (short)0, c, false, false);
            V8F D; D.v = c;

            // D layout: lane<16 -> rows 0..7, col=lane; lane>=16 -> rows 8..15
            float colv = 0.0f;
#pragma unroll
            for (int e = 0; e < 8; ++e) {
                float v = EXP2(D.f[e]);           // exp(cos / tau)
                racc[e] += v;
                colv    += v;
            }
            // All 32 lanes atomic-add; lanes l and l+16 share a column slot
            atomicAdd(&colacc[jj * 16 + (lane & 15)], colv);   // ds_add_f32

            b0 = nb0;
            b1 = nb1;
        }

        // Row partial sums: butterfly-reduce over each half-wave (16 columns)
        // with single-instruction ds_swizzle xor steps, then one global f32
        // atomic per row (NCHUNK contributors per row).
        const int rbase = itile * 16 + ((lane & 16) ? 8 : 0);
#pragma unroll
        for (int e = 0; e < 8; ++e) {
            float v = racc[e];
            v += SWZ_XOR(v, 1);
            v += SWZ_XOR(v, 2);
            v += SWZ_XOR(v, 4);
            v += SWZ_XOR(v, 8);
            if ((lane & 15) == 0) unsafeAtomicAdd(&rowsum[rbase + e], v);
        }
    }

    __syncthreads();
    // Flush per-block column partials: one hardware f32 atomic per column
    const int cbase = jbase * 16;
    for (int i = threadIdx.x; i < CCOLS; i += 256) {
        float v = colacc[i];
        if (v != 0.0f) unsafeAtomicAdd(&colsum[cbase + i], v);
    }
}

// ---------------------------------------------------------------------------
// Kernel 3: scalar loss.
// loss = mean_i[ 0.5*log(rowsum_i+eps) + 0.5*log(colsum_i+eps) - cos(i,i)/tau ]
// U1 rows carry TAU_INV*log2(e); multiply the raw dot by ln2 to recover
// cos/tau for the diagonal term.
// ---------------------------------------------------------------------------
__global__ __launch_bounds__(1024) void finalize_kernel(
    const float* __restrict__ U1, const float* __restrict__ U2,
    const float* __restrict__ rowsum, const float* __restrict__ colsum,
    float* __restrict__ out)
{
    __shared__ float red[1024];
    float acc = 0.0f;
    for (int i = threadIdx.x; i < NROWS; i += 1024) {
        float d = 0.0f;
#pragma unroll
        for (int k = 0; k < 8; ++k)
            d += U1[(size_t)i * 8 + k] * U2[(size_t)i * 8 + k];
        acc += 0.5f * __logf(rowsum[i] + LOSS_EPS)
             + 0.5f * __logf(colsum[i] + LOSS_EPS)
             - d * LN2F;
    }
    red[threadIdx.x] = acc;
    __syncthreads();
#pragma unroll
    for (int s = 512; s > 0; s >>= 1) {
        if (threadIdx.x < (unsigned)s) red[threadIdx.x] += red[threadIdx.x + s];
        __syncthreads();
    }
    if (threadIdx.x == 0) out[0] = red[0] / (float)NROWS;
}

// ---------------------------------------------------------------------------
// Host-side launcher (graph-capture safe: only kernel launches on `stream`)
// ---------------------------------------------------------------------------
extern "C" void kernel_launch(void* const* d_in, const int* in_sizes, int n_in,
                              void* d_out, int out_size, void* d_ws, size_t ws_size,
                              hipStream_t stream) {
    const float* z_mp = (const float*)d_in[0];
    const float* z_sc = (const float*)d_in[1];
    const float* W1   = (const float*)d_in[2];
    const float* b1   = (const float*)d_in[3];
    const float* W2   = (const float*)d_in[4];
    const float* b2   = (const float*)d_in[5];

    // Workspace layout (floats, total ~720 KB):
    //   U1f:    [0,       80000)   normalized rows * TAU_INV*log2(e)
    //   U2f:    [80000,  160000)   normalized rows
    //   pad:    [160000, 160128)   one zero tile (prefetch target)
    //   rowsum: [160128, 170128)
    //   colsum: [170128, 180128)
    float* U1f    = (float*)d_ws;
    float* U2f    = U1f + (size_t)NROWS * 8;
    float* tail   = U2f + (size_t)NROWS * 8;       // pad + rowsum + colsum
    float* rowsum = tail + 128;
    float* colsum = rowsum + NROWS;
    const int tail_n = 128 + 2 * NROWS;            // zero pad/rowsum/colsum in one go

    zero_kernel<<<(tail_n + 255) / 256, 256, 0, stream>>>(tail, tail_n);
    proj_norm_kernel<<<(2 * NROWS + 255) / 256, 256, 0, stream>>>(
        z_mp, z_sc, W1, b1, W2, b2, U1f, U2f);
    sim_tile_kernel<<<dim3((NT + 7) / 8, NCHUNK), 256, 0, stream>>>(
        U1f, U2f, rowsum, colsum, JCHUNK);
    finalize_kernel<<<1, 1024, 0, stream>>>(U1f, U2f, rowsum, colsum, (float*)d_out);
}